// Ctc_Loss_53051436040493
// MI455X (gfx1250) — compile-verified
//
#include <hip/hip_runtime.h>
#include <math.h>

#define NEGV (-1e30f)

typedef __attribute__((ext_vector_type(4))) unsigned int u32x4;
typedef __attribute__((ext_vector_type(8))) int          i32x8;
typedef __attribute__((ext_vector_type(4))) int          i32x4;

// ---------------------------------------------------------------------------
// TDM: DMA one contiguous row of `nelem` fp32 from global memory into LDS.
// Descriptor per CDNA5 ISA ch. 8 (D# groups 0/1); 2-D tile with tile_dim1=1.
// ---------------------------------------------------------------------------
__device__ __forceinline__ void tdm_load_row_f32(unsigned int lds_byte_addr,
                                                 const void* gptr,
                                                 unsigned int nelem) {
  unsigned long long ga = (unsigned long long)(size_t)gptr;

  u32x4 g0;
  g0.x = 1u;                                        // count=1, user descriptor
  g0.y = lds_byte_addr;                             // lds_addr [63:32]
  g0.z = (unsigned int)(ga & 0xFFFFFFFFu);          // global_addr lo32
  g0.w = (unsigned int)((ga >> 32) & 0x01FFFFFFu)   // global_addr [56:32]
         | 0x80000000u;                              // type=2 ("image")

  i32x8 g1;
  g1[0] = (int)(2u << 16);                          // data_size=2 (4B), mask=0
  g1[1] = (int)((nelem & 0xFFFFu) << 16);           // tensor_dim0 lo16
  g1[2] = (int)((nelem >> 16) | (1u << 16));        // tensor_dim0 hi16 | tensor_dim1=1
  g1[3] = (int)((nelem & 0xFFFFu) << 16);           // tensor_dim1 hi16=0 | tile_dim0
  g1[4] = 1;                                        // tile_dim1=1, tile_dim2=0
  g1[5] = (int)nelem;                               // tensor_dim0_stride lo32
  g1[6] = 0;                                        // stride hi16 | dim1_stride lo16
  g1[7] = 0;

  i32x4 z4 = {0, 0, 0, 0};                          // groups 2/3 unused (2-D)
#if defined(__clang_major__) && (__clang_major__ >= 23)
  i32x8 z8 = {0, 0, 0, 0, 0, 0, 0, 0};
  __builtin_amdgcn_tensor_load_to_lds(g0, g1, z4, z4, z8, 0);
#else
  __builtin_amdgcn_tensor_load_to_lds(g0, g1, z4, z4, 0);
#endif
}

__device__ __forceinline__ float lae3(float x, float y, float z) {
  float m = fmaxf(fmaxf(x, y), z);
  return m + __logf(__expf(x - m) + __expf(y - m) + __expf(z - m));
}
__device__ __forceinline__ float lae2(float x, float y) {
  float m = fmaxf(x, y);
  return m + __logf(__expf(x - m) + __expf(y - m));
}

// ---------------------------------------------------------------------------
// Kernel 1: one wave per (n,t) row. TDM row -> LDS, logsumexp over V=4000
// with b128 LDS reads (row = 1000 float4: 31 uniform rounds + 8-float4 tail),
// gather 65 extended-label log-probs -> lpe[n][t][65].
// ---------------------------------------------------------------------------
__global__ __launch_bounds__(32)
void ctc_rows(const float* __restrict__ yp, const int* __restrict__ tgt,
              float* __restrict__ lpe) {
  __shared__ __align__(16) float sh[4000];
  const int row  = blockIdx.x;        // n*T + t, T=256
  const int n    = row >> 8;
  const int lane = threadIdx.x;

  unsigned int lds_base = (unsigned int)(size_t)(void*)&sh[0];
  tdm_load_row_f32(lds_base, yp + (size_t)row * 4000u, 4000u);
  __builtin_amdgcn_s_wait_tensorcnt(0);
  asm volatile("" ::: "memory");       // LDS now holds the row (DMA-written)

  const float4* sh4 = reinterpret_cast<const float4*>(sh);   // 1000 float4

  // ---- pass 1: row max (uniform trip count; tail predicated, branchless)
  float m = -INFINITY;
#pragma unroll 4
  for (int k = 0; k < 31; ++k) {                             // ds_load_b128
    const float4 v = sh4[(k << 5) + lane];
    m = fmaxf(m, fmaxf(fmaxf(v.x, v.y), fmaxf(v.z, v.w)));
  }
  {
    const float4 v = sh4[992 + (lane & 7)];
    const float t = fmaxf(fmaxf(v.x, v.y), fmaxf(v.z, v.w));
    m = (lane < 8) ? fmaxf(m, t) : m;
  }
#pragma unroll
  for (int off = 16; off; off >>= 1) m = fmaxf(m, __shfl_xor(m, off, 32));

  // ---- pass 2: sum exp(x - m)
  float s = 0.f;
#pragma unroll 4
  for (int k = 0; k < 31; ++k) {                             // ds_load_b128
    const float4 v = sh4[(k << 5) + lane];
    s += (__expf(v.x - m) + __expf(v.y - m)) +
         (__expf(v.z - m) + __expf(v.w - m));
  }
  {
    const float4 v = sh4[992 + (lane & 7)];
    const float e = (__expf(v.x - m) + __expf(v.y - m)) +
                    (__expf(v.z - m) + __expf(v.w - m));
    s += (lane < 8) ? e : 0.f;
  }
#pragma unroll
  for (int off = 16; off; off >>= 1) s += __shfl_xor(s, off, 32);

  const float norm = m + __logf(s);    // log_softmax = x - norm

  const int* tg = tgt + n * 32;        // S = 32
  float* dst = lpe + (size_t)row * 65u;
  const int l0 = lane, l1 = lane + 32;
  const int e0 = (l0 & 1) ? tg[(l0 - 1) >> 1] : 0;   // ext label (blank=0)
  const int e1 = (l1 & 1) ? tg[(l1 - 1) >> 1] : 0;
  dst[l0] = sh[e0] - norm;
  dst[l1] = sh[e1] - norm;
  if (lane == 0) dst[64] = sh[0] - norm;
}

// ---------------------------------------------------------------------------
// Kernel 2: one wave per sample. 65 DP states packed as lane, lane+32, and
// state 64 on lane 0. Neighbor alphas via wave32 shuffles — no barriers.
// ---------------------------------------------------------------------------
__global__ __launch_bounds__(32)
void ctc_dp(const float* __restrict__ lpe, const int* __restrict__ tgt,
            float* __restrict__ nll) {
  const int n = blockIdx.x, lane = threadIdx.x;
  const int* tg = tgt + n * 32;
  const int s1i = lane + 32;

  // skip (s-2) allowed only for odd states l>=3 with label != label two back
  const bool k0 = (lane & 1) && (lane >= 3) &&
                  (tg[(lane - 1) >> 1] != tg[(lane - 3) >> 1]);
  const bool k1 = (s1i & 1) &&
                  (tg[(s1i - 1) >> 1] != tg[(s1i - 3) >> 1]);

  const float* lp = lpe + (size_t)n * 256u * 65u;
  float a0 = (lane == 0) ? lp[0] : ((lane == 1) ? lp[1] : NEGV);
  float a1 = NEGV, a2 = NEGV;

  for (int t = 1; t < 256; ++t) {
    const float* lt = lp + t * 65;
    const float p0 = lt[lane];
    const float p1 = lt[s1i];
    const float p2 = (lane == 0) ? lt[64] : 0.f;

    float a0m1 = __shfl_up(a0, 1);      // alpha[lane-1]
    float a0m2 = __shfl_up(a0, 2);      // alpha[lane-2]
    float a1m1 = __shfl_up(a1, 1);      // alpha[lane+31]
    float a1m2 = __shfl_up(a1, 2);      // alpha[lane+30]
    const float a0_30 = __shfl(a0, 30);
    const float a0_31 = __shfl(a0, 31);
    const float a1_31 = __shfl(a1, 31); // alpha[63]
    if (lane == 0) { a0m1 = NEGV; a0m2 = NEGV; a1m1 = a0_31; a1m2 = a0_30; }
    if (lane == 1) { a0m2 = NEGV; a1m2 = a0_31; }

    const float b0 = lae3(a0, a0m1, k0 ? a0m2 : NEGV) + p0;
    const float b1 = lae3(a1, a1m1, k1 ? a1m2 : NEGV) + p1;
    const float b2 = (lane == 0) ? (lae2(a2, a1_31) + p2) : NEGV; // state 64 blank
    a0 = b0; a1 = b1; a2 = b2;
  }
  const float a63 = __shfl(a1, 31);
  if (lane == 0) nll[n] = -lae2(a2, a63);
}

// ---------------------------------------------------------------------------
// Kernel 3: mean(nll / S) over N=64, S=32.
// ---------------------------------------------------------------------------
__global__ __launch_bounds__(32)
void ctc_reduce(const float* __restrict__ nll, float* __restrict__ out) {
  const int lane = threadIdx.x;
  float s = nll[lane] + nll[lane + 32];
#pragma unroll
  for (int off = 16; off; off >>= 1) s += __shfl_xor(s, off, 32);
  if (lane == 0) out[0] = s * (1.0f / (64.0f * 32.0f));
}

extern "C" void kernel_launch(void* const* d_in, const int* in_sizes, int n_in,
                              void* d_out, int out_size, void* d_ws, size_t ws_size,
                              hipStream_t stream) {
  const float* yp  = (const float*)d_in[0];   // [64, 256, 4000] fp32
  const int*   tgt = (const int*)d_in[1];     // [64, 32] int32
  float* lpe = (float*)d_ws;                              // 64*256*65 fp32
  float* nll = lpe + (size_t)64 * 256 * 65;               // 64 fp32

  ctc_rows  <<<64 * 256, 32, 0, stream>>>(yp, tgt, lpe);
  ctc_dp    <<<64,       32, 0, stream>>>(lpe, tgt, nll);
  ctc_reduce<<<1,        32, 0, stream>>>(nll, (float*)d_out);
}